// SAGE_72258529788632
// MI455X (gfx1250) — compile-verified
//
#include <hip/hip_runtime.h>
#include <hip/hip_bf16.h>

typedef __attribute__((ext_vector_type(2))) float v2f;
typedef __attribute__((ext_vector_type(8))) float v8f;

#define N_NODES 100000
#define N_EDGES 1000000
#define IN_F    64
#define HID_F   64
#define OUT_F   32

// ---------------------------------------------------------------------------
// Zero a float buffer (grid-stride).
// ---------------------------------------------------------------------------
__global__ void zero_f32(float* __restrict__ p, long n) {
  long i = (long)blockIdx.x * blockDim.x + threadIdx.x;
  long stride = (long)gridDim.x * blockDim.x;
  for (; i < n; i += stride) p[i] = 0.0f;
}

// ---------------------------------------------------------------------------
// Edge scatter: agg[dst] += w * feat[src]  (and cnt[dst] += 1 on first pass).
// One thread per (edge, 4-feature group): coalesced float4 gather from the
// L2-resident feature table, 4 global_atomic_add_f32 into L2-resident agg.
// ---------------------------------------------------------------------------
__global__ void edge_scatter(const float* __restrict__ feat,
                             const long long* __restrict__ ei,
                             const float* __restrict__ ea,
                             float* __restrict__ agg,
                             float* cnt, int nEdges) {
  int t = blockIdx.x * blockDim.x + threadIdx.x;
  int e = t >> 4;   // 16 groups of 4 features = 64 features
  int g = t & 15;
  if (e >= nEdges) return;
  int src = (int)ei[e];
  int dst = (int)ei[nEdges + e];
  float w = ea[e];
  const float4* f4 = (const float4*)(feat + (long)src * 64);
  float4 v = f4[g];
  float* out = agg + (long)dst * 64 + g * 4;
  atomicAdd(out + 0, v.x * w);
  atomicAdd(out + 1, v.y * w);
  atomicAdd(out + 2, v.z * w);
  atomicAdd(out + 3, v.w * w);
  if (cnt != nullptr && g == 0) atomicAdd(cnt + dst, 1.0f);
}

// ---------------------------------------------------------------------------
// Fused GraphConv dense stage, one 16x16 output tile per wave32:
//   out[m,n] = sum_k (agg[m,k]/max(cnt[m],1)) * Wrel[n,k]
//            + sum_k  xin[m,k]               * Wroot[n,k]
//            + brel[n]            (then optional sigmoid)
// K = 64 fixed. Uses V_WMMA_F32_16X16X4_F32 (full fp32 — GEMM is compute-
// light here, so keep reference precision).
//
// Fragment layouts per CDNA5 ISA 7.12.2 (32-bit, wave32):
//   A 16x4 : lanes 0-15 -> M=lane, {v.x=K0, v.y=K1}; lanes 16-31 -> {K2,K3}
//   B 4x16 : lanes 0-15 -> N=lane, {v.x=K0, v.y=K1}; lanes 16-31 -> {K2,K3}
//   C/D    : VGPR j -> row j (lanes 0-15) / row j+8 (lanes 16-31), col = lane%16
// ---------------------------------------------------------------------------
__global__ void gemm_fused_wmma(const float* __restrict__ agg,
                                const float* __restrict__ cnt,
                                const float* __restrict__ xin,
                                const float* __restrict__ Wrel,
                                const float* __restrict__ brel,
                                const float* __restrict__ Wroot,
                                float* __restrict__ out,
                                int rowTiles, int colTiles, int outF,
                                int applySigmoid) {
  int wavesPerBlock = blockDim.x >> 5;
  int gw = blockIdx.x * wavesPerBlock + ((int)threadIdx.x >> 5);
  if (gw >= rowTiles * colTiles) return;   // wave-uniform: EXEC stays all-1s
  int rt = gw / colTiles;
  int ct = gw - rt * colTiles;

  int lane = threadIdx.x & 31;
  int half = lane >> 4;      // 0: K pair {0,1}, 1: K pair {2,3}
  int l    = lane & 15;

  int row = rt * 16 + l;     // A-fragment row for this lane
  int n   = ct * 16 + l;     // B-fragment column for this lane

  float rcp = 1.0f / fmaxf(cnt[row], 1.0f);   // mean aggregation

  const float* aggRow   = agg   + (long)row * 64;
  const float* xRow     = xin   + (long)row * 64;
  const float* wrelRow  = Wrel  + (long)n * 64;   // W[n][k] -> B[k][n]
  const float* wrootRow = Wroot + (long)n * 64;

  v8f acc = {};
  #pragma unroll
  for (int kk = 0; kk < 16; ++kk) {         // K = 64, 4 per WMMA
    int kA = kk * 4 + half * 2;             // this lane's K pair

    v2f a = *(const v2f*)(aggRow + kA);
    a *= rcp;
    v2f b = *(const v2f*)(wrelRow + kA);
    acc = __builtin_amdgcn_wmma_f32_16x16x4_f32(
        false, a, false, b, (short)0, acc, false, false);

    v2f a2 = *(const v2f*)(xRow + kA);
    v2f b2 = *(const v2f*)(wrootRow + kA);
    acc = __builtin_amdgcn_wmma_f32_16x16x4_f32(
        false, a2, false, b2, (short)0, acc, false, false);
  }

  float bias = brel[n];
  int colOut = ct * 16 + l;
  #pragma unroll
  for (int j = 0; j < 8; ++j) {
    int orow = rt * 16 + j + half * 8;
    float v = acc[j] + bias;
    if (applySigmoid) v = 1.0f / (1.0f + __expf(-v));
    out[(long)orow * outF + colOut] = v;
  }
}

// ---------------------------------------------------------------------------
// Host-side launcher. Workspace layout (floats):
//   agg : N_NODES*64   (reused for both layers)
//   cnt : N_NODES      (computed once; same dst distribution both layers)
//   h   : N_NODES*64   (layer-1 activations)
// ---------------------------------------------------------------------------
extern "C" void kernel_launch(void* const* d_in, const int* in_sizes, int n_in,
                              void* d_out, int out_size, void* d_ws, size_t ws_size,
                              hipStream_t stream) {
  const float*     x      = (const float*)d_in[0];
  const long long* ei     = (const long long*)d_in[1];   // int64 edge_index
  const float*     ea     = (const float*)d_in[2];
  const float*     Wrel1  = (const float*)d_in[3];
  const float*     brel1  = (const float*)d_in[4];
  const float*     Wroot1 = (const float*)d_in[5];
  const float*     Wrel2  = (const float*)d_in[6];
  const float*     brel2  = (const float*)d_in[7];
  const float*     Wroot2 = (const float*)d_in[8];
  float* out = (float*)d_out;

  float* agg = (float*)d_ws;
  float* cnt = agg + (long)N_NODES * 64;
  float* h   = cnt + N_NODES;

  const long aggN = (long)N_NODES * 64;
  const int  scatterThreads = N_EDGES * 16;
  const int  scatterBlocks  = (scatterThreads + 255) / 256;

  // Layer 1
  zero_f32<<<2048, 256, 0, stream>>>(agg, aggN + N_NODES);   // agg + cnt contiguous
  edge_scatter<<<scatterBlocks, 256, 0, stream>>>(x, ei, ea, agg, cnt, N_EDGES);
  {
    int rowTiles = N_NODES / 16, colTiles = HID_F / 16;      // 6250 x 4
    int waves = rowTiles * colTiles;
    gemm_fused_wmma<<<(waves + 7) / 8, 256, 0, stream>>>(
        agg, cnt, x, Wrel1, brel1, Wroot1, h, rowTiles, colTiles, HID_F, 1);
  }

  // Layer 2 (reuse agg buffer)
  zero_f32<<<2048, 256, 0, stream>>>(agg, aggN);
  edge_scatter<<<scatterBlocks, 256, 0, stream>>>(h, ei, ea, agg, nullptr, N_EDGES);
  {
    int rowTiles = N_NODES / 16, colTiles = OUT_F / 16;      // 6250 x 2
    int waves = rowTiles * colTiles;
    gemm_fused_wmma<<<(waves + 7) / 8, 256, 0, stream>>>(
        agg, cnt, h, Wrel2, brel2, Wroot2, out, rowTiles, colTiles, OUT_F, 0);
  }
}